// SpatialPyramidPooling3D_73418170957951
// MI455X (gfx1250) — compile-verified
//
#include <hip/hip_runtime.h>
#include <math.h>

// Spatial Pyramid Pooling 3D, pool_list = {1,2,4}
// x: [B=4, H=64, W=64, D=64, C=64] f32 (channels-last)
// out: [B, 73*C] = [4, 4672] f32, segments: p=1 @0 (64), p=2 @64 (512), p=4 @576 (4096)
//
// Memory-bound: 256 MB single-pass read -> ~11 us floor at 23.3 TB/s.
// Strategy: compute 4x4x4 grid of 16^3-block channel maxima (one pass,
// coalesced 128-bit nontemporal loads), fold into p4/p2/p1 output slots
// with global_atomic_max_num_f32 (associative/commutative -> deterministic).

typedef float v4f __attribute__((ext_vector_type(4)));

#define SPP_C 64
#define SPP_S 64
#define SPP_OUT_PER_B 4672  // 64 + 512 + 4096

__global__ void spp3d_init_kernel(float* __restrict__ out, int n) {
    int i = blockIdx.x * blockDim.x + threadIdx.x;
    if (i < n) out[i] = -__builtin_inff();
}

__global__ __launch_bounds__(256) void spp3d_main_kernel(
        const float* __restrict__ x, float* __restrict__ out) {
    // Workgroup id -> (batch, block_i, block_j, block_k, h-slab)
    const int bx   = blockIdx.x;
    const int b    = bx >> 8;          // 4 batches
    const int rem  = bx & 255;         // 64 blocks * 4 slabs
    const int bi   = (rem >> 6) & 3;   // block index along H (16-voxel blocks)
    const int bj   = (rem >> 4) & 3;   // along W
    const int bk   = (rem >> 2) & 3;   // along D
    const int slab = rem & 3;          // 4-row h-slab within the block

    const int tid = threadIdx.x;
    const int c4  = tid & 15;          // channel quad: channels [c4*4, c4*4+4)
    const int s   = tid >> 4;          // d offset within block (0..15)

    const size_t strideH = (size_t)SPP_S * SPP_S * SPP_C;  // 64*64*64
    const size_t strideW = (size_t)SPP_S * SPP_C;          // 64*64
    const float* xb = x + (size_t)b * SPP_S * strideH;

    const int h0 = bi * 16 + slab * 4;
    const int w0 = bj * 16;
    const int d  = bk * 16 + s;

    v4f acc;
    acc[0] = acc[1] = acc[2] = acc[3] = -__builtin_inff();

    // Each thread streams 4 (h) x 16 (w) voxels of its (d, channel-quad).
    // Per wave: lanes cover 512 contiguous bytes -> full-width coalescing.
    for (int hh = 0; hh < 4; ++hh) {
        const float* basep = xb + (size_t)(h0 + hh) * strideH
                                + (size_t)w0 * strideW
                                + (size_t)d * SPP_C
                                + (size_t)(c4 * 4);
        #pragma unroll
        for (int ww = 0; ww < 16; ++ww) {
            // Streamed once, no reuse, working set > L2 -> nontemporal b128 load
            v4f v = __builtin_nontemporal_load(
                reinterpret_cast<const v4f*>(basep + (size_t)ww * strideW));
            acc[0] = fmaxf(acc[0], v[0]);
            acc[1] = fmaxf(acc[1], v[1]);
            acc[2] = fmaxf(acc[2], v[2]);
            acc[3] = fmaxf(acc[3], v[3]);
        }
    }

    // Cross-thread (over s = d-offset) max via LDS tree, 4 KB
    __shared__ v4f red[16][16];  // [s][c4]
    red[s][c4] = acc;
    __syncthreads();
    #pragma unroll
    for (int stride = 8; stride >= 1; stride >>= 1) {
        if (s < stride) {
            v4f o = red[s + stride][c4];
            v4f m = red[s][c4];
            m[0] = fmaxf(m[0], o[0]);
            m[1] = fmaxf(m[1], o[1]);
            m[2] = fmaxf(m[2], o[2]);
            m[3] = fmaxf(m[3], o[3]);
            red[s][c4] = m;
        }
        __syncthreads();
    }

    if (s == 0) {
        v4f m = red[0][c4];
        float* ob = out + (size_t)b * SPP_OUT_PER_B;
        const int c0 = c4 * 4;
        // p=4 segment: region (bi,bj,bk) row-major
        const int p4off = 576 + (((bi * 4) + bj) * 4 + bk) * SPP_C + c0;
        // p=2 segment: region (bi/2, bj/2, bk/2) row-major
        const int p2off = 64 + ((((bi >> 1) * 2) + (bj >> 1)) * 2 + (bk >> 1)) * SPP_C + c0;
        #pragma unroll
        for (int q = 0; q < 4; ++q) {
            atomicMax(&ob[p4off + q], m[q]);   // 4 slab-contributors
            atomicMax(&ob[p2off + q], m[q]);   // 32 contributors
            atomicMax(&ob[c0 + q],    m[q]);   // p=1: 256 contributors
        }
    }
}

extern "C" void kernel_launch(void* const* d_in, const int* in_sizes, int n_in,
                              void* d_out, int out_size, void* d_ws, size_t ws_size,
                              hipStream_t stream) {
    (void)in_sizes; (void)n_in; (void)d_ws; (void)ws_size;
    const float* x = (const float*)d_in[0];
    float* out = (float*)d_out;

    // 1) out := -inf (atomic-max identity; harness poisons d_out)
    spp3d_init_kernel<<<(out_size + 255) / 256, 256, 0, stream>>>(out, out_size);

    // 2) one-pass streaming block-max + hierarchical fold via f32 max atomics
    //    grid = 4 batches * 64 blocks * 4 slabs = 1024 WGs of 256 threads (8 waves)
    spp3d_main_kernel<<<4 * 256, 256, 0, stream>>>(x, out);
}